// Net_2791728742833
// MI455X (gfx1250) — compile-verified
//
#include <hip/hip_runtime.h>

#define NNODES 100000
#define NEDGES 1600000
#define FH 64

typedef __attribute__((ext_vector_type(2))) float v2f;
typedef __attribute__((ext_vector_type(8))) float v8f;

// ---------------------------------------------------------------------------
// Degree / normalization
// ---------------------------------------------------------------------------
__global__ void k_init_deg(float* deg, int n) {
    int i = blockIdx.x * blockDim.x + threadIdx.x;
    if (i < n) deg[i] = 1.0f;   // self-loop counted in degree
}

__global__ void k_deg_accum(const int* __restrict__ ei, float* deg, int e) {
    int i = blockIdx.x * blockDim.x + threadIdx.x;
    if (i < e) atomicAdd(&deg[ei[e + i]], 1.0f);   // dst = ei[E + i]
}

__global__ void k_rsqrt_inplace(float* deg, int n) {
    int i = blockIdx.x * blockDim.x + threadIdx.x;
    if (i < n) deg[i] = rsqrtf(deg[i]);
}

// ---------------------------------------------------------------------------
// WMMA fp32 GEMM:  C[M x 64] = A[M x K] * B[K x 64]
// One wave per 16-row slab, computing all 64 output columns (4 accumulators)
// so the A slab is loaded once and fed to 4 V_WMMA_F32_16X16X4_F32 per chunk.
// K=10 tail uses clamped-index loads + mask-multiply (no EXEC divergence).
// M must be a multiple of 16 (100000 = 6250*16).
// ---------------------------------------------------------------------------
template <int K>
__global__ void k_gemm_wmma64(const float* __restrict__ A,
                              const float* __restrict__ B,
                              float* __restrict__ C, int M) {
    constexpr int NCOL = 64;
    constexpr int KP = (K + 3) & ~3;

    int gtid = blockIdx.x * blockDim.x + threadIdx.x;
    int wave = gtid >> 5;               // one wave per 16-row slab
    int lane = gtid & 31;
    int m0 = wave * 16;
    if (m0 >= M) return;                // wave-uniform: EXEC all-1s at WMMAs

    int mr = lane & 15;                 // A row within slab / B column base
    int kg = (lane >> 4) << 1;          // 0 for lanes 0-15, 2 for lanes 16-31

    const float* Arow = A + (size_t)(m0 + mr) * K;

    v8f acc0 = {}, acc1 = {}, acc2 = {}, acc3 = {};
#pragma unroll
    for (int k0 = 0; k0 < KP; k0 += 4) {
        int ka = k0 + kg;
        constexpr bool FULL_TAIL = (K % 4 == 0);
        bool full = FULL_TAIL || (k0 + 4 <= K);     // compile-time per iteration

        v2f a;
        v2f b0, b1, b2, b3;
        if (full) {
            a.x = Arow[ka];
            a.y = Arow[ka + 1];
            const float* B0 = B + (size_t)ka * NCOL + mr;
            const float* B1 = B + (size_t)(ka + 1) * NCOL + mr;
            b0.x = B0[0];  b0.y = B1[0];
            b1.x = B0[16]; b1.y = B1[16];
            b2.x = B0[32]; b2.y = B1[32];
            b3.x = B0[48]; b3.y = B1[48];
        } else {
            // clamp indices in-bounds, zero via mask multiply (finite data)
            int kc0 = (ka     < K) ? ka     : 0;
            int kc1 = (ka + 1 < K) ? ka + 1 : 0;
            float msk0 = (ka     < K) ? 1.0f : 0.0f;
            float msk1 = (ka + 1 < K) ? 1.0f : 0.0f;
            a.x = Arow[kc0] * msk0;
            a.y = Arow[kc1] * msk1;
            const float* B0 = B + (size_t)kc0 * NCOL + mr;
            const float* B1 = B + (size_t)kc1 * NCOL + mr;
            b0.x = B0[0]  * msk0;  b0.y = B1[0]  * msk1;
            b1.x = B0[16] * msk0;  b1.y = B1[16] * msk1;
            b2.x = B0[32] * msk0;  b2.y = B1[32] * msk1;
            b3.x = B0[48] * msk0;  b3.y = B1[48] * msk1;
        }
        acc0 = __builtin_amdgcn_wmma_f32_16x16x4_f32(false, a, false, b0,
                                                     (short)0, acc0, false, false);
        acc1 = __builtin_amdgcn_wmma_f32_16x16x4_f32(false, a, false, b1,
                                                     (short)0, acc1, false, false);
        acc2 = __builtin_amdgcn_wmma_f32_16x16x4_f32(false, a, false, b2,
                                                     (short)0, acc2, false, false);
        acc3 = __builtin_amdgcn_wmma_f32_16x16x4_f32(false, a, false, b3,
                                                     (short)0, acc3, false, false);
    }

    // C/D layout: VGPR v -> row m0+v (lanes 0-15) or m0+v+8 (lanes 16-31)
    int col   = lane & 15;
    int rbase = m0 + ((lane >> 4) << 3);
#pragma unroll
    for (int v = 0; v < 8; ++v) {
        float* crow = C + (size_t)(rbase + v) * NCOL + col;
        crow[0]  = acc0[v];
        crow[16] = acc1[v];
        crow[32] = acc2[v];
        crow[48] = acc3[v];
    }
}

// ---------------------------------------------------------------------------
// Edge aggregation, F=64: agg[dst] += h[src] * dinv[src]*dinv[dst]
// 16 threads per edge, float4 gather + 4 fp32 global atomics each.
// ---------------------------------------------------------------------------
__global__ void k_edge_agg64(const int* __restrict__ ei,
                             const float* __restrict__ h,
                             const float* __restrict__ dinv,
                             float* __restrict__ agg) {
    long long t = (long long)blockIdx.x * blockDim.x + threadIdx.x;
    if (t >= (long long)NEDGES * 16) return;
    int e  = (int)(t >> 4);
    int f4 = (int)(t & 15);
    int s = ei[e];
    int d = ei[NEDGES + e];
    float nrm = dinv[s] * dinv[d];
    float4 v = *((const float4*)(h + (size_t)s * FH) + f4);
    float* out = agg + (size_t)d * FH + f4 * 4;
    atomicAdd(out + 0, v.x * nrm);
    atomicAdd(out + 1, v.y * nrm);
    atomicAdd(out + 2, v.z * nrm);
    atomicAdd(out + 3, v.w * nrm);
}

// Edge aggregation, F=4 (one thread per edge)
__global__ void k_edge_agg4(const int* __restrict__ ei,
                            const float* __restrict__ h4,
                            const float* __restrict__ dinv,
                            float* __restrict__ agg) {
    int e = blockIdx.x * blockDim.x + threadIdx.x;
    if (e >= NEDGES) return;
    int s = ei[e];
    int d = ei[NEDGES + e];
    float nrm = dinv[s] * dinv[d];
    float4 v = ((const float4*)h4)[s];
    float* out = agg + (size_t)d * 4;
    atomicAdd(out + 0, v.x * nrm);
    atomicAdd(out + 1, v.y * nrm);
    atomicAdd(out + 2, v.z * nrm);
    atomicAdd(out + 3, v.w * nrm);
}

// ---------------------------------------------------------------------------
// h[i] = relu(agg[i] + h[i]*dinv[node]^2 + bias[f])   (in-place on h)
// ---------------------------------------------------------------------------
__global__ void k_self_bias_relu(float* __restrict__ h,
                                 const float* __restrict__ agg,
                                 const float* __restrict__ dinv,
                                 const float* __restrict__ bias,
                                 int total) {
    int i = blockIdx.x * blockDim.x + threadIdx.x;
    if (i >= total) return;
    int node = i >> 6;          // / 64
    int f    = i & 63;
    float dv = dinv[node];
    float v  = agg[i] + h[i] * dv * dv + bias[f];
    h[i] = fmaxf(v, 0.0f);
}

// ---------------------------------------------------------------------------
// Layer-3 GEMM: h3[N x 4] = h[N x 64] * W3[64 x 4]   (tiny, scalar per node)
// ---------------------------------------------------------------------------
__global__ void k_gemm3(const float* __restrict__ h,
                        const float* __restrict__ W3,
                        float* __restrict__ h3, int n) {
    int i = blockIdx.x * blockDim.x + threadIdx.x;
    if (i >= n) return;
    const float* row = h + (size_t)i * FH;
    float a0 = 0.f, a1 = 0.f, a2 = 0.f, a3 = 0.f;
#pragma unroll
    for (int k = 0; k < FH; ++k) {
        float hv = row[k];
        a0 = fmaf(hv, W3[k * 4 + 0], a0);
        a1 = fmaf(hv, W3[k * 4 + 1], a1);
        a2 = fmaf(hv, W3[k * 4 + 2], a2);
        a3 = fmaf(hv, W3[k * 4 + 3], a3);
    }
    float4 r = make_float4(a0, a1, a2, a3);
    ((float4*)h3)[i] = r;
}

// ---------------------------------------------------------------------------
// Final: z = agg3 + h3*dinv^2 + b3 ; out = log_softmax(z) over 4 classes
// ---------------------------------------------------------------------------
__global__ void k_final(const float* __restrict__ agg3,
                        const float* __restrict__ h3,
                        const float* __restrict__ dinv,
                        const float* __restrict__ b3,
                        float* __restrict__ out, int n) {
    int i = blockIdx.x * blockDim.x + threadIdx.x;
    if (i >= n) return;
    float dv = dinv[i];
    float d2 = dv * dv;
    float4 a = ((const float4*)agg3)[i];
    float4 h = ((const float4*)h3)[i];
    float z0 = a.x + h.x * d2 + b3[0];
    float z1 = a.y + h.y * d2 + b3[1];
    float z2 = a.z + h.z * d2 + b3[2];
    float z3 = a.w + h.w * d2 + b3[3];
    float m  = fmaxf(fmaxf(z0, z1), fmaxf(z2, z3));
    float s  = __expf(z0 - m) + __expf(z1 - m) + __expf(z2 - m) + __expf(z3 - m);
    float l  = m + __logf(s);
    float4 r = make_float4(z0 - l, z1 - l, z2 - l, z3 - l);
    ((float4*)out)[i] = r;
}

// ---------------------------------------------------------------------------
extern "C" void kernel_launch(void* const* d_in, const int* in_sizes, int n_in,
                              void* d_out, int out_size, void* d_ws, size_t ws_size,
                              hipStream_t stream) {
    const float* x  = (const float*)d_in[0];   // [N,10]
    const int*   ei = (const int*)d_in[1];     // [2,E] flat: src then dst
    const float* W1 = (const float*)d_in[2];   // [10,64]
    const float* b1 = (const float*)d_in[3];   // [64]
    const float* W2 = (const float*)d_in[4];   // [64,64]
    const float* b2 = (const float*)d_in[5];   // [64]
    const float* W3 = (const float*)d_in[6];   // [64,4]
    const float* b3 = (const float*)d_in[7];   // [4]
    float* out = (float*)d_out;                // [N,4]

    // workspace partition
    float* dinv = (float*)d_ws;                         // N
    float* bufA = dinv + NNODES;                        // N*64
    float* bufB = bufA + (size_t)NNODES * FH;           // N*64
    float* h3   = bufB + (size_t)NNODES * FH;           // N*4
    float* agg3 = h3   + (size_t)NNODES * 4;            // N*4

    const int TB = 256;
    auto blk = [](long long n) { return (int)((n + 255) / 256); };

    // --- degree / normalization ---
    k_init_deg<<<blk(NNODES), TB, 0, stream>>>(dinv, NNODES);
    k_deg_accum<<<blk(NEDGES), TB, 0, stream>>>(ei, dinv, NEDGES);
    k_rsqrt_inplace<<<blk(NNODES), TB, 0, stream>>>(dinv, NNODES);

    // one wave per 16-row slab
    const long long gemmThreads = (long long)(NNODES / 16) * 32;   // 200000

    // --- layer 1: h1 = x @ W1 -> bufA ; aggregate -> bufB ; fuse into bufA ---
    k_gemm_wmma64<10><<<blk(gemmThreads), TB, 0, stream>>>(x, W1, bufA, NNODES);
    hipMemsetAsync(bufB, 0, (size_t)NNODES * FH * sizeof(float), stream);
    k_edge_agg64<<<blk((long long)NEDGES * 16), TB, 0, stream>>>(ei, bufA, dinv, bufB);
    k_self_bias_relu<<<blk((long long)NNODES * FH), TB, 0, stream>>>(bufA, bufB, dinv, b1, NNODES * FH);

    // --- layer 2: h2 = h1 @ W2 -> bufB ; aggregate -> bufA ; fuse into bufB ---
    k_gemm_wmma64<64><<<blk(gemmThreads), TB, 0, stream>>>(bufA, W2, bufB, NNODES);
    hipMemsetAsync(bufA, 0, (size_t)NNODES * FH * sizeof(float), stream);
    k_edge_agg64<<<blk((long long)NEDGES * 16), TB, 0, stream>>>(ei, bufB, dinv, bufA);
    k_self_bias_relu<<<blk((long long)NNODES * FH), TB, 0, stream>>>(bufB, bufA, dinv, b2, NNODES * FH);

    // --- layer 3: h3 = h2 @ W3 ; aggregate F=4 ; log_softmax ---
    k_gemm3<<<blk(NNODES), TB, 0, stream>>>(bufB, W3, h3, NNODES);
    hipMemsetAsync(agg3, 0, (size_t)NNODES * 4 * sizeof(float), stream);
    k_edge_agg4<<<blk(NEDGES), TB, 0, stream>>>(ei, h3, dinv, agg3);
    k_final<<<blk(NNODES), TB, 0, stream>>>(agg3, h3, dinv, b3, out, NNODES);
}